// WeaveGather_74732430950922
// MI455X (gfx1250) — compile-verified
//
#include <hip/hip_runtime.h>
#include <cmath>

#define N_ATOMS 131072
#define N_MOLS  4096
#define DFEAT   128
#define NK      11
#define DK      (DFEAT * NK)   // 1408
#define NDEPTH  128

typedef __attribute__((ext_vector_type(2))) float v2f;
typedef __attribute__((ext_vector_type(8))) float v8f;

struct MemConsts {
    float mean[NK];
    float inv_std[NK];
    float c[NK];       // 0.5 / (-log(std*sqrt(2*pi)))  (positive)
};

__device__ __forceinline__ int lower_bound_i32(const int* __restrict__ a, int n, int val) {
    int lo = 0, hi = n;
    while (lo < hi) {
        int mid = (lo + hi) >> 1;
        if (a[mid] < val) lo = mid + 1; else hi = mid;
    }
    return lo;
}

// ---------------------------------------------------------------------------
// Kernel 1: fused membership expansion + segment sum.
// grid = N_MOLS blocks, block = DFEAT(128) threads (4 waves).
// Thread d of block m accumulates mols[m, d*11 .. d*11+10] over the (sorted,
// contiguous) atom range belonging to molecule m. x is read exactly once,
// fully coalesced (512B per atom row). No atomics needed.
// ---------------------------------------------------------------------------
__global__ void membership_segsum_kernel(const float* __restrict__ x,
                                         const int* __restrict__ atom_split,
                                         float* __restrict__ mols,
                                         MemConsts mc) {
    const int mol = blockIdx.x;
    const int d   = threadIdx.x;          // 0..127

    const int start = lower_bound_i32(atom_split, N_ATOMS, mol);
    const int end   = lower_bound_i32(atom_split, N_ATOMS, mol + 1);

    float acc[NK];
#pragma unroll
    for (int k = 0; k < NK; ++k) acc[k] = 0.0f;

    for (int a = start; a < end; ++a) {
        const float v = x[(size_t)a * DFEAT + d];
        float o[NK];
        float s = 0.0f;
#pragma unroll
        for (int k = 0; k < NK; ++k) {
            const float z  = (v - mc.mean[k]) * mc.inv_std[k];
            const float ok = fmaf(-mc.c[k], z * z, 1.0f);   // logp / (-log_norm)
            o[k] = ok;
            s += ok;
        }
        const float r = 1.0f / (s + 1e-7f);
#pragma unroll
        for (int k = 0; k < NK; ++k)
            acc[k] = fmaf(o[k] + 1e-7f, r, acc[k]);
    }

    const size_t base = (size_t)mol * DK + (size_t)d * NK;
#pragma unroll
    for (int k = 0; k < NK; ++k) mols[base + k] = acc[k];
}

// ---------------------------------------------------------------------------
// Kernel 2: out = tanh(mols @ W + b) using V_WMMA_F32_16X16X4_F32 (full f32).
// grid = N_MOLS/16 blocks of one wave32. Each wave owns one 16-row M tile and
// computes all 8 N tiles (N=128): 8 independent accumulators per k-step for
// WMMA latency hiding, A fragment reused 8x.
//
// Fragment layouts per CDNA5 ISA 7.12.2:
//   A 16x4 f32 : lane l (l<16): M=l, K={k0,k0+1}; lane l+16: M=l, K={k0+2,k0+3}
//   B 4x16 f32 : VGPR j, lanes 0-15: K=k0+j, N=lane; lanes 16-31: K=k0+2+j
//   C/D 16x16  : VGPR v, lanes 0-15: M=v, N=lane; lanes 16-31: M=v+8, N=lane-16
// ---------------------------------------------------------------------------
__global__ void gemm_bias_tanh_wmma(const float* __restrict__ A,     // [N_MOLS, DK]
                                    const float* __restrict__ W,     // [DK, NDEPTH]
                                    const float* __restrict__ bias,  // [NDEPTH]
                                    float* __restrict__ out) {       // [N_MOLS, NDEPTH]
    const int lane  = threadIdx.x;        // 0..31
    const int mTile = blockIdx.x;         // 0..255
    const int half  = lane >> 4;          // 0 or 1 (selects K pair)
    const int l15   = lane & 15;

    v8f acc[8];
#pragma unroll
    for (int t = 0; t < 8; ++t)
#pragma unroll
        for (int i = 0; i < 8; ++i) acc[t][i] = 0.0f;

    float bv[8];
#pragma unroll
    for (int t = 0; t < 8; ++t) bv[t] = bias[t * 16 + l15];

    // A fragment base: row (mTile*16 + l15), starting at K offset half*2.
    const float* arow = A + (size_t)(mTile * 16 + l15) * DK + half * 2;

    for (int k0 = 0; k0 < DK; k0 += 4) {
        // 8-byte aligned: k0 % 4 == 0 and row stride DK=1408 is even.
        const v2f a = *(const v2f*)(arow + k0);
#pragma unroll
        for (int t = 0; t < 8; ++t) {
            const float* wp = W + (size_t)(k0 + half * 2) * NDEPTH + t * 16 + l15;
            v2f bfrag;
            bfrag.x = wp[0];
            bfrag.y = wp[NDEPTH];
            acc[t] = __builtin_amdgcn_wmma_f32_16x16x4_f32(
                /*neg_a=*/false, a, /*neg_b=*/false, bfrag,
                /*c_mod=*/(short)0, acc[t], /*reuse_a=*/false, /*reuse_b=*/false);
        }
    }

    const int rowBase = mTile * 16 + (half << 3);
#pragma unroll
    for (int t = 0; t < 8; ++t) {
#pragma unroll
        for (int v = 0; v < 8; ++v) {
            const float val = tanhf(acc[t][v] + bv[t]);
            out[(size_t)(rowBase + v) * NDEPTH + t * 16 + l15] = val;
        }
    }
}

// ---------------------------------------------------------------------------
extern "C" void kernel_launch(void* const* d_in, const int* in_sizes, int n_in,
                              void* d_out, int out_size, void* d_ws, size_t ws_size,
                              hipStream_t stream) {
    const float* x          = (const float*)d_in[0];
    const int*   atom_split = (const int*)d_in[1];
    const float* W          = (const float*)d_in[2];
    const float* b          = (const float*)d_in[3];
    float*       out        = (float*)d_out;
    float*       mols       = (float*)d_ws;   // 4096*1408*4 = 23.1 MB scratch

    static const double MEANS[NK] = {-1.645, -1.08, -0.739, -0.468, -0.228, 0.0,
                                      0.228,  0.468, 0.739,  1.08,   1.645};
    static const double STDS[NK]  = {0.283, 0.17, 0.134, 0.118, 0.114, 0.114,
                                     0.114, 0.118, 0.134, 0.17,  0.283};
    MemConsts mc;
    for (int k = 0; k < NK; ++k) {
        const double ln = std::log(STDS[k] * 2.5066282746310002); // log(std*sqrt(2pi)) < 0
        mc.mean[k]    = (float)MEANS[k];
        mc.inv_std[k] = (float)(1.0 / STDS[k]);
        mc.c[k]       = (float)(0.5 / (-ln));
    }

    membership_segsum_kernel<<<N_MOLS, DFEAT, 0, stream>>>(x, atom_split, mols, mc);
    gemm_bias_tanh_wmma<<<N_MOLS / 16, 32, 0, stream>>>(mols, W, b, out);
}